// Repro_87402584474058
// MI455X (gfx1250) — compile-verified
//
#include <hip/hip_runtime.h>
#include <math.h>

// ---------------------------------------------------------------------------
// Fused pipeline for MI455X (gfx1250, wave32), one workgroup per batch b:
//   - M_b = p1[b] + 0.975*p3[b] built ONCE into LDS (256KB of the 320KB WGP
//     LDS), reused by all three stage GEMMs; per-stage scatter patch is a
//     single LDS store.
//   - 3x [12,256]x[256,256] GEMM via V_WMMA_F32_16X16X4_F32, A & B fragments
//     from LDS, per-column sinusoidal gain folded into the D-store.
//   - rDFT (ortho) via two WMMA GEMMs with on-the-fly __sincosf twiddles.
//   - z = imag @ p19[b] via WMMA; out = (real - z)/1.15.
// ~50 MFLOP / ~5 MB total -> L2-resident and latency bound; LDS reuse of M_b
// cuts global traffic 3x. Async global->LDS (ASYNCcnt path) used for the
// activation load.
// ---------------------------------------------------------------------------

typedef __attribute__((ext_vector_type(2))) float v2f;
typedef __attribute__((ext_vector_type(8))) float v8f;

#define NN   256
#define HH   12
#define NF   129
#define XS   264   // row stride for 16x256 stage buffers (bank-conflict pad)
#define FS   148   // row stride for 16x144 spectral buffers
#define MS   256   // row stride for M in LDS
#define PI_F 3.14159265358979323846f

#if defined(__has_builtin)
#if __has_builtin(__builtin_amdgcn_global_load_async_to_lds_b32)
#define HAVE_ASYNC_LDS 1
#endif
#endif

typedef __attribute__((address_space(1))) int* gptr_i32;
typedef __attribute__((address_space(3))) int* lptr_i32;

// dynamic LDS layout (floats): M | bufA | bufB | gmul | Rb | Ib
#define SM_M    0
#define SM_BUFA (SM_M + NN * MS)
#define SM_BUFB (SM_BUFA + 16 * XS)
#define SM_GM   (SM_BUFB + 16 * XS)
#define SM_RB   (SM_GM + NN)
#define SM_IB   (SM_RB + 16 * FS)
#define SM_TOT  (SM_IB + 16 * FS)

__global__ __launch_bounds__(256) void fused_pipeline_kernel(
    const float* __restrict__ p1,  const float* __restrict__ p2,
    const float* __restrict__ p3,
    const float* __restrict__ t0,  const float* __restrict__ t1,
    const float* __restrict__ t2,
    const int*   __restrict__ p5,  const int*   __restrict__ p6,
    const int*   __restrict__ p7,  const int*   __restrict__ p8,
    const float* __restrict__ f0,  const float* __restrict__ ph0,
    const float* __restrict__ f1,  const float* __restrict__ ph1,
    const float* __restrict__ f2,  const float* __restrict__ ph2,
    const float* __restrict__ p19, float* __restrict__ out)
{
    extern __shared__ float smem[];
    float* Mlds = smem + SM_M;
    float* bufA = smem + SM_BUFA;
    float* bufB = smem + SM_BUFB;
    float* gmul = smem + SM_GM;
    float* Rb   = smem + SM_RB;
    float* Ib   = smem + SM_IB;

    const int b    = blockIdx.x;
    const int tid  = threadIdx.x;
    const int lane = tid & 31;
    const int wv   = tid >> 5;

    const float* p1b  = p1  + b * NN * NN;
    const float* p3b  = p3  + b * NN * NN;
    const float* p19b = p19 + b * NF * NF;

    // locate the scatter patch row for this batch: gathered = table[p5,p6]
    int p6r = -1, p7r = 0, p8r = 0;
    for (int i = 0; i < 6; ++i)
        if (p5[i] == b) { p6r = p6[i]; p7r = p7[i]; p8r = p8[i]; }

    // ---- build M_b = p1[b] + 0.975*p3[b] once into LDS (coalesced b128) ----
    {
        const float4* p1v = (const float4*)p1b;
        const float4* p3v = (const float4*)p3b;
        float4*       mv  = (float4*)Mlds;
        for (int i = tid; i < (NN * NN) / 4; i += 256) {
            float4 a = p1v[i], c = p3v[i], r;
            r.x = fmaf(0.975f, c.x, a.x);
            r.y = fmaf(0.975f, c.y, a.y);
            r.z = fmaf(0.975f, c.z, a.z);
            r.w = fmaf(0.975f, c.w, a.w);
            mv[i] = r;
        }
    }

    // ---- load x_b = p2[:, b, :] into LDS (async global->LDS, ASYNCcnt) ----
#if HAVE_ASYNC_LDS
    for (int idx = tid; idx < HH * NN; idx += 256) {
        const int h = idx >> 8, d = idx & 255;
        __builtin_amdgcn_global_load_async_to_lds_b32(
            (gptr_i32)(p2 + (h * 6 + b) * NN + d),
            (lptr_i32)(bufA + h * XS + d),
            0, 0);
    }
#else
    for (int idx = tid; idx < HH * NN; idx += 256) {
        const int h = idx >> 8, d = idx & 255;
        bufA[h * XS + d] = p2[(h * 6 + b) * NN + d];
    }
#endif
    for (int idx = tid; idx < 4 * NN; idx += 256) {     // zero-pad rows 12..15
        const int h = HH + (idx >> 8), d = idx & 255;
        bufA[h * XS + d] = 0.0f;
    }

    // warm L2->L1 for the final z-GEMM operand
    for (int i = tid; i * 64 < NF * NF; i += 256)
        __builtin_prefetch(p19b + i * 64, 0, 1);

#if HAVE_ASYNC_LDS
#if __has_builtin(__builtin_amdgcn_s_wait_asynccnt)
    __builtin_amdgcn_s_wait_asynccnt(0);
#else
    asm volatile("s_wait_asynccnt 0x0" ::: "memory");
#endif
#endif
    __syncthreads();

    float* cur = bufA;
    float* nxt = bufB;

    const float* tabs[3] = { t0, t1, t2 };
    const float* frs[3]  = { f0, f1, f2 };
    const float* phs[3]  = { ph0, ph1, ph2 };

    const int m  = lane & 15;   // A row / B-C-D column within tile
    const int kh = lane >> 4;   // K-half selector

    // ---------------- three chained GEMM stages ----------------
    for (int s = 0; s < 3; ++s) {
        // per-column gain: g = sin(2*pi*e*f + ph)^2*0.1 + 0.95; stage1 mul else div
        {
            const float fr = frs[s][0], ph = phs[s][0];
            const float sn = __sinf(2.0f * PI_F * (float)tid * fr + ph);
            const float g  = sn * sn * 0.1f + 0.95f;
            gmul[tid] = (s == 1) ? g : 1.0f / g;
        }
        // per-stage scatter patch: one LDS store into M
        if (tid == 0 && p6r >= 0)
            Mlds[p7r * MS + p8r] = tabs[s][b * NN + p6r]
                                 + 0.975f * p3b[p7r * NN + p8r];
        __syncthreads();

        // y = x @ M   (M resident in LDS, reused by all 3 stages)
        for (int t = 0; t < 2; ++t) {
            const int n0 = (wv + t * 8) * 16;
            const int n  = n0 + m;
            v8f c = {};
            for (int k0 = 0; k0 < NN; k0 += 4) {
                v2f a, bb;
                a.x  = cur[m * XS + k0 + 2 * kh];
                a.y  = cur[m * XS + k0 + 2 * kh + 1];
                bb.x = Mlds[(k0 + kh) * MS + n];        // B vgpr0: K=2*0+kh
                bb.y = Mlds[(k0 + 2 + kh) * MS + n];    // B vgpr1: K=2*1+kh
                c = __builtin_amdgcn_wmma_f32_16x16x4_f32(
                        false, a, false, bb, (short)0, c, false, false);
            }
            // D layout: vgpr r -> rows r / r+8; gain folded into the store
            const float gm = gmul[n];
            const int rowoff = (lane < 16) ? 0 : 8;
            #pragma unroll
            for (int r = 0; r < 8; ++r)
                nxt[(r + rowoff) * XS + n] = c[r] * gm;
        }
        __syncthreads();

        float* tmp = cur; cur = nxt; nxt = tmp;
    }

    // ---------------- rfft (ortho): WMMA against on-the-fly cos/sin ----------
    {
        const float w0 = 2.0f * PI_F / 256.0f;
        for (int nt = wv; nt < 9; nt += 8) {
            const int kf = nt * 16 + m;      // frequency bin (0..143, valid <129)
            v8f cR = {}, cI = {};
            for (int k0 = 0; k0 < NN; k0 += 4) {
                const int d0 = k0 + kh;
                const int d1 = k0 + 2 + kh;
                v2f a, bc, bs;
                a.x = cur[m * XS + k0 + 2 * kh];
                a.y = cur[m * XS + k0 + 2 * kh + 1];
                float s0, c0, s1, c1;
                __sincosf(w0 * (float)d0 * (float)kf, &s0, &c0);
                __sincosf(w0 * (float)d1 * (float)kf, &s1, &c1);
                bc.x =  c0 * 0.0625f;  bc.y =  c1 * 0.0625f;   // 1/sqrt(256)
                bs.x = -s0 * 0.0625f;  bs.y = -s1 * 0.0625f;
                cR = __builtin_amdgcn_wmma_f32_16x16x4_f32(
                         false, a, false, bc, (short)0, cR, false, false);
                cI = __builtin_amdgcn_wmma_f32_16x16x4_f32(
                         false, a, false, bs, (short)0, cI, false, false);
            }
            const int rowoff = (lane < 16) ? 0 : 8;
            #pragma unroll
            for (int r = 0; r < 8; ++r) {
                Rb[(r + rowoff) * FS + kf] = cR[r];
                Ib[(r + rowoff) * FS + kf] = (kf < NF) ? cI[r] : 0.0f; // zero K-pad
            }
        }
    }
    __syncthreads();

    // ---------------- z = imag @ p19[b], out = (real - z)/1.15 ---------------
    {
        for (int nt = wv; nt < 9; nt += 8) {
            const int e = nt * 16 + m;
            v8f cz = {};
            for (int k0 = 0; k0 < 132; k0 += 4) {   // K padded 129 -> 132
                const int d0 = k0 + kh;
                const int d1 = k0 + 2 + kh;
                v2f a, bb;
                a.x = Ib[m * FS + k0 + 2 * kh];
                a.y = Ib[m * FS + k0 + 2 * kh + 1];
                bb.x = (d0 < NF && e < NF) ? p19b[d0 * NF + e] : 0.0f;
                bb.y = (d1 < NF && e < NF) ? p19b[d1 * NF + e] : 0.0f;
                cz = __builtin_amdgcn_wmma_f32_16x16x4_f32(
                         false, a, false, bb, (short)0, cz, false, false);
            }
            const int rowoff = (lane < 16) ? 0 : 8;
            #pragma unroll
            for (int r = 0; r < 8; ++r) {
                const int row = r + rowoff;
                if (row < HH && e < NF)
                    out[(row * 6 + b) * NF + e] =
                        (Rb[row * FS + e] - cz[r]) * (1.0f / 1.15f);
            }
        }
    }
}

extern "C" void kernel_launch(void* const* d_in, const int* in_sizes, int n_in,
                              void* d_out, int out_size, void* d_ws, size_t ws_size,
                              hipStream_t stream) {
    const float* p1  = (const float*)d_in[0];   // [6,256,256]
    const float* p2  = (const float*)d_in[1];   // [12,6,256]
    const float* p3  = (const float*)d_in[2];   // [6,256,256]
    const float* t0  = (const float*)d_in[3];   // [6,256]
    const int*   p5  = (const int*)  d_in[4];   // [6]
    const int*   p6  = (const int*)  d_in[5];   // [6]
    const int*   p7  = (const int*)  d_in[6];   // [6]
    const int*   p8  = (const int*)  d_in[7];   // [6]
    const float* f0  = (const float*)d_in[8];   // scalar
    const float* ph0 = (const float*)d_in[9];   // scalar
    const float* t1  = (const float*)d_in[10];  // [6,256]
    const float* f1  = (const float*)d_in[11];  // scalar
    const float* ph1 = (const float*)d_in[12];  // scalar
    const float* t2  = (const float*)d_in[13];  // [6,256]
    const float* f2  = (const float*)d_in[14];  // scalar
    const float* ph2 = (const float*)d_in[15];  // scalar
    const float* p19 = (const float*)d_in[16];  // [6,129,129]

    const int B = in_sizes[4];                  // 6 batches -> 6 workgroups
    const size_t shbytes = (size_t)SM_TOT * sizeof(float);  // ~316 KB of 320 KB WGP LDS
    fused_pipeline_kernel<<<dim3(B), dim3(256), shbytes, stream>>>(
        p1, p2, p3, t0, t1, t2, p5, p6, p7, p8,
        f0, ph0, f1, ph1, f2, ph2, p19, (float*)d_out);
}